// GCNEncoder_80642305950441
// MI455X (gfx1250) — compile-verified
//
#include <hip/hip_runtime.h>
#include <hip/hip_bf16.h>

#define NN 50000      // nodes
#define NE 800000     // edges
#define F  128        // feature dim (in == hid)
#define NG 512        // graphs
#define OUTW 384      // 3*F

typedef __attribute__((ext_vector_type(2))) float v2f;
typedef __attribute__((ext_vector_type(8))) float v8f;

// ---------------------------------------------------------------- degree
__global__ __launch_bounds__(256) void k_deg_init(float* deg) {
    int i = blockIdx.x * 256 + threadIdx.x;
    if (i < NN) deg[i] = 1.0f;                    // self-loop
}

__global__ __launch_bounds__(256) void k_deg_edges(const int* __restrict__ ei, float* deg) {
    int e = blockIdx.x * 256 + threadIdx.x;
    if (e < NE) atomicAdd(&deg[ei[NE + e]], 1.0f);
}

__global__ __launch_bounds__(256) void k_rsqrt(float* deg) {
    int i = blockIdx.x * 256 + threadIdx.x;
    if (i < NN) deg[i] = rsqrtf(deg[i]);          // in-place -> dinv
}

// ---------------------------------------------------------------- GEMM (WMMA fp32)
// H[M,F] = (relu?)(A[M,F]) @ W[F,F].  One block = 16-row tile, 8 waves = 8 col tiles.
// A frag (16x4 f32, 2 VGPR): lane L, vgpr j -> m = L&15, k = 2*(L>>4)+j
// B frag (4x16 f32, 2 VGPR): lane L, vgpr j -> k = 2*(L>>4)+j, n = L&15
// C/D   (16x16 f32, 8 VGPR): lane L, vgpr r -> m = r + 8*(L>>4), n = L&15
template <bool RELU_IN>
__global__ __launch_bounds__(256) void k_gemm_wmma(const float* __restrict__ A,
                                                   const float* __restrict__ W,
                                                   float* __restrict__ H) {
    __shared__ float sA[16 * F];                  // 8 KB, shared by all 8 waves
    const int tid  = threadIdx.x;
    const int row0 = blockIdx.x * 16;             // NN == 3125*16 exactly

    for (int idx = tid; idx < 16 * F; idx += 256) {
        float v = A[(size_t)(row0 + (idx >> 7)) * F + (idx & (F - 1))];
        if (RELU_IN) v = fmaxf(v, 0.0f);
        sA[idx] = v;
    }
    __syncthreads();

    const int wave  = tid >> 5;                   // col tile 0..7
    const int lane  = tid & 31;
    const int lhalf = lane >> 4;                  // 0/1
    const int l16   = lane & 15;
    const int col0  = wave * 16;

    v8f acc = {};
    #pragma unroll 4
    for (int kk = 0; kk < F; kk += 4) {
        const int kf = kk + 2 * lhalf;
        v2f a, b;
        a.x = sA[l16 * F + kf];
        a.y = sA[l16 * F + kf + 1];
        b.x = W[(size_t)kf * F + col0 + l16];
        b.y = W[(size_t)(kf + 1) * F + col0 + l16];
        acc = __builtin_amdgcn_wmma_f32_16x16x4_f32(false, a, false, b,
                                                    (short)0, acc, false, false);
    }

    #pragma unroll
    for (int r = 0; r < 8; ++r) {
        const int m = r + 8 * lhalf;
        H[(size_t)(row0 + m) * F + col0 + l16] = acc[r];
    }
}

// ---------------------------------------------------------------- aggregation
// agg[i][f] = h[i][f]*dinv[i]^2 + b[f]   (self-loop + bias, no atomics needed)
__global__ __launch_bounds__(256) void k_agg_init(const float* __restrict__ h,
                                                  const float* __restrict__ dinv,
                                                  const float* __restrict__ bias,
                                                  float* __restrict__ agg) {
    size_t idx = (size_t)blockIdx.x * 256 + threadIdx.x;
    if (idx >= (size_t)NN * F) return;
    int i = (int)(idx >> 7), f = (int)(idx & (F - 1));
    float d = dinv[i];
    agg[idx] = h[idx] * d * d + bias[f];
}

// one wave per edge: gather h[src], scale by dinv[src]*dinv[dst], atomic-add into agg[dst]
__global__ __launch_bounds__(256) void k_edge_scatter(const int* __restrict__ ei,
                                                      const float* __restrict__ dinv,
                                                      const float* __restrict__ h,
                                                      float* __restrict__ agg) {
    int e = blockIdx.x * 8 + (threadIdx.x >> 5);
    if (e >= NE) return;
    const int lane = threadIdx.x & 31;
    const int src  = ei[e];
    const int dst  = ei[NE + e];
    const float norm = dinv[src] * dinv[dst];
    const float4 hv = ((const float4*)(h + (size_t)src * F))[lane];
    float* o = agg + (size_t)dst * F + lane * 4;
    atomicAdd(o + 0, hv.x * norm);
    atomicAdd(o + 1, hv.y * norm);
    atomicAdd(o + 2, hv.z * norm);
    atomicAdd(o + 3, hv.w * norm);
}

// ---------------------------------------------------------------- pooling
__global__ __launch_bounds__(256) void k_zero_out(float* out, float* cnt) {
    int i = blockIdx.x * 256 + threadIdx.x;
    if (i < NG * OUTW) out[i] = 0.0f;
    if (i < NG) cnt[i] = 0.0f;
}

// one wave per node; relu fused at read; sum + bitwise-max (values >= 0) into d_out
__global__ __launch_bounds__(256) void k_pool(const float* __restrict__ agg,
                                              const int* __restrict__ batch,
                                              float* __restrict__ out,
                                              float* __restrict__ cnt) {
    int i = blockIdx.x * 8 + (threadIdx.x >> 5);
    if (i >= NN) return;
    const int lane = threadIdx.x & 31;
    const int g = batch[i];
    float4 v = ((const float4*)(agg + (size_t)i * F))[lane];
    v.x = fmaxf(v.x, 0.0f); v.y = fmaxf(v.y, 0.0f);
    v.z = fmaxf(v.z, 0.0f); v.w = fmaxf(v.w, 0.0f);
    float* s = out + (size_t)g * OUTW + lane * 4;
    atomicAdd(s + 0, v.x);
    atomicAdd(s + 1, v.y);
    atomicAdd(s + 2, v.z);
    atomicAdd(s + 3, v.w);
    unsigned int* mx = (unsigned int*)(out + (size_t)g * OUTW + F + lane * 4);
    atomicMax(mx + 0, __float_as_uint(v.x));      // monotone for floats >= 0
    atomicMax(mx + 1, __float_as_uint(v.y));
    atomicMax(mx + 2, __float_as_uint(v.z));
    atomicMax(mx + 3, __float_as_uint(v.w));
    if (lane == 0) atomicAdd(&cnt[g], 1.0f);
}

__global__ __launch_bounds__(256) void k_mean(float* __restrict__ out,
                                              const float* __restrict__ cnt) {
    int idx = blockIdx.x * 256 + threadIdx.x;
    if (idx >= NG * F) return;
    int g = idx >> 7, f = idx & (F - 1);
    out[(size_t)g * OUTW + 2 * F + f] =
        out[(size_t)g * OUTW + f] / fmaxf(cnt[g], 1.0f);
}

// ---------------------------------------------------------------- driver
extern "C" void kernel_launch(void* const* d_in, const int* in_sizes, int n_in,
                              void* d_out, int out_size, void* d_ws, size_t ws_size,
                              hipStream_t stream) {
    const float* x     = (const float*)d_in[0];
    const float* W1    = (const float*)d_in[1];
    const float* b1    = (const float*)d_in[2];
    const float* W2    = (const float*)d_in[3];
    const float* b2    = (const float*)d_in[4];
    const int*   ei    = (const int*)d_in[5];
    const int*   batch = (const int*)d_in[6];
    float* out = (float*)d_out;

    char* ws = (char*)d_ws;
    const size_t featBytes = (size_t)NN * F * sizeof(float);   // 25.6 MB
    float* dinv = (float*)ws;                                  // 200 KB slot
    float* h    = (float*)(ws + (256 << 10));
    float* agg  = (float*)(ws + (256 << 10) + featBytes);
    float* cnt  = (float*)(ws + (256 << 10) + 2 * featBytes);

    const int nodeBlocks = (NN + 255) / 256;       // 196
    const int edgeBlocks = (NE + 255) / 256;       // 3125
    const int gemmBlocks = NN / 16;                // 3125
    const int elemBlocks = (NN * F) / 256;         // 25000
    const int scatBlocks = NE / 8;                 // 100000
    const int poolBlocks = NN / 8;                 // 6250

    // degree / normalization
    k_deg_init <<<nodeBlocks, 256, 0, stream>>>(dinv);
    k_deg_edges<<<edgeBlocks, 256, 0, stream>>>(ei, dinv);
    k_rsqrt    <<<nodeBlocks, 256, 0, stream>>>(dinv);

    // layer 1
    k_gemm_wmma<false><<<gemmBlocks, 256, 0, stream>>>(x, W1, h);
    k_agg_init    <<<elemBlocks, 256, 0, stream>>>(h, dinv, b1, agg);
    k_edge_scatter<<<scatBlocks, 256, 0, stream>>>(ei, dinv, h, agg);

    // layer 2 (relu fused into GEMM A-stage); h buffer reused
    k_gemm_wmma<true><<<gemmBlocks, 256, 0, stream>>>(agg, W2, h);
    k_agg_init    <<<elemBlocks, 256, 0, stream>>>(h, dinv, b2, agg);
    k_edge_scatter<<<scatBlocks, 256, 0, stream>>>(ei, dinv, h, agg);

    // readout (relu fused into pooling read)
    k_zero_out<<<(NG * OUTW + 255) / 256, 256, 0, stream>>>(out, cnt);
    k_pool    <<<poolBlocks, 256, 0, stream>>>(agg, batch, out, cnt);
    k_mean    <<<(NG * F + 255) / 256, 256, 0, stream>>>(out, cnt);
}